// RotateNMS_81080392614230
// MI455X (gfx1250) — compile-verified
//
#include <hip/hip_runtime.h>
#include <stdint.h>

#define NB     2048
#define TOPN   1000
#define WORDS  64          // 2048 bits / 32
#define THR    0.7f

typedef float v2f __attribute__((ext_vector_type(2)));
typedef float v8f __attribute__((ext_vector_type(8)));

// ---------------------------------------------------------------------------
// Exact convex intersection area of two rotated rectangles (Sutherland-Hodgman)
// ca, cb: 4 CCW corners as x0,y0,...,x3,y3.
// __noinline__ keeps exactly ONE copy of this large routine (I-cache).
// ---------------------------------------------------------------------------
__device__ __noinline__ float interArea(const float* __restrict__ ca,
                                        const float* __restrict__ cb) {
  float px[12], py[12], qx[12], qy[12];
  int n = 4;
#pragma unroll
  for (int k = 0; k < 4; ++k) { px[k] = ca[2 * k]; py[k] = ca[2 * k + 1]; }
#pragma unroll 1
  for (int e = 0; e < 4; ++e) {
    const float axp = cb[2 * e], ayp = cb[2 * e + 1];
    const float bxp = cb[(2 * e + 2) & 7], byp = cb[(2 * e + 3) & 7];
    const float ex = bxp - axp, ey = byp - ayp;
    int m = 0;
    float sx = px[n - 1], sy = py[n - 1];
    float ds = ex * (sy - ayp) - ey * (sx - axp);
#pragma unroll 1
    for (int k = 0; k < n; ++k) {
      const float cx = px[k], cy = py[k];
      const float dc = ex * (cy - ayp) - ey * (cx - axp);
      const bool cin = (dc >= 0.0f), pin = (ds >= 0.0f);
      if (cin != pin) {                      // crossing: emit intersection
        const float t = ds / (ds - dc);      // denom strictly nonzero here
        qx[m] = sx + t * (cx - sx);
        qy[m] = sy + t * (cy - sy);
        ++m;
      }
      if (cin) { qx[m] = cx; qy[m] = cy; ++m; }
      sx = cx; sy = cy; ds = dc;
    }
    n = m;
    if (n == 0) return 0.0f;
#pragma unroll 1
    for (int k = 0; k < n; ++k) { px[k] = qx[k]; py[k] = qy[k]; }
  }
  if (n < 3) return 0.0f;
  float acc = 0.0f;
  float sx = px[n - 1], sy = py[n - 1];
#pragma unroll 1
  for (int k = 0; k < n; ++k) {
    acc += sx * py[k] - px[k] * sy;
    sx = px[k]; sy = py[k];
  }
  return 0.5f * fabsf(acc);
}

// ---------------------------------------------------------------------------
// Kernel 1: per-box precompute (corners, |center|^2, bounding radius, area)
// ---------------------------------------------------------------------------
__global__ void prep_kernel(const float* __restrict__ boxes,
                            float* __restrict__ corners,
                            float* __restrict__ nsq,
                            float* __restrict__ rad,
                            float* __restrict__ area) {
  const int i = blockIdx.x * blockDim.x + threadIdx.x;
  if (i >= NB) return;
  const float xc = boxes[i * 5 + 0];
  const float yc = boxes[i * 5 + 1];
  const float w  = boxes[i * 5 + 2];
  const float h  = boxes[i * 5 + 3];
  const float t  = boxes[i * 5 + 4];
  float c, s;
  __sincosf(t, &s, &c);
  const float lx[4] = {-0.5f * w, 0.5f * w, 0.5f * w, -0.5f * w};
  const float ly[4] = {-0.5f * h, -0.5f * h, 0.5f * h, 0.5f * h};
#pragma unroll
  for (int k = 0; k < 4; ++k) {
    corners[i * 8 + 2 * k + 0] = xc + lx[k] * c - ly[k] * s;
    corners[i * 8 + 2 * k + 1] = yc + lx[k] * s + ly[k] * c;
  }
  nsq[i]  = xc * xc + yc * yc;
  rad[i]  = 0.5f * sqrtf(w * w + h * h);
  area[i] = w * h;
}

// ---------------------------------------------------------------------------
// Kernel 2: pairwise overlap bitmask.  One wave owns a 16x32 pair tile.
// WMMA f32 16x16x4 computes all 512 center dot-products -> circle-test
// prefilter; exact polygon clip only for surviving pairs (compact ctz loop);
// ballot-compacted full mask words, no atomics.
// ---------------------------------------------------------------------------
__global__ void pair_kernel(const float* __restrict__ boxes,
                            const float* __restrict__ corners,
                            const float* __restrict__ nsq,
                            const float* __restrict__ rad,
                            const float* __restrict__ area,
                            unsigned* __restrict__ mask) {
  const int lane = threadIdx.x & 31;
  const int wid  = (blockIdx.x * blockDim.x + threadIdx.x) >> 5;
  const int ti   = wid >> 6;   // 0..127  (16-row tile)
  const int tj   = wid & 63;   // 0..63   (32-col word tile)
  const int i0   = ti << 4;
  const int j0   = tj << 5;
  const int la   = lane & 15;

  // A fragment: 16x4 (K=0,1 real; K=2,3 zero-pad in lanes 16-31).
  // B fragments: 4x16 (one per 16-col half of the word tile).
  float axv = 0.f, ayv = 0.f, b0x = 0.f, b0y = 0.f, b1x = 0.f, b1y = 0.f;
  if (lane < 16) {
    axv = boxes[(i0 + la) * 5 + 0];
    ayv = boxes[(i0 + la) * 5 + 1];
    b0x = boxes[(j0 + la) * 5 + 0];
    b0y = boxes[(j0 + la) * 5 + 1];
    b1x = boxes[(j0 + 16 + la) * 5 + 0];
    b1y = boxes[(j0 + 16 + la) * 5 + 1];
  }
  v2f A  = {axv, ayv};
  v2f B0 = {b0x, b0y};
  v2f B1 = {b1x, b1y};
  v8f C  = {0.f, 0.f, 0.f, 0.f, 0.f, 0.f, 0.f, 0.f};
  // D[m][n] = p_i(m) . p_j(n)
  v8f D0 = __builtin_amdgcn_wmma_f32_16x16x4_f32(false, A, false, B0,
                                                 (short)0, C, false, false);
  v8f D1 = __builtin_amdgcn_wmma_f32_16x16x4_f32(false, A, false, B1,
                                                 (short)0, C, false, false);

  // This lane's two columns (same for both wave halves).
  const int col0 = j0 + la;
  const int col1 = col0 + 16;
  const float nj0 = nsq[col0], rj0 = rad[col0], aj0 = area[col0];
  const float nj1 = nsq[col1], rj1 = rad[col1], aj1 = area[col1];
  __builtin_prefetch(&corners[col0 * 8], 0, 1);
  __builtin_prefetch(&corners[col1 * 8], 0, 1);

  const int rbase = i0 + ((lane >> 4) << 3);  // +8 for upper wave half

  // --- cheap circle prefilter, fully unrolled (constant D[v] extraction) ---
  unsigned pmask = 0u;  // bit (v + 8*tile): pair may overlap
#pragma unroll
  for (int v = 0; v < 8; ++v) {
    const int row = rbase + v;
    const float ni = nsq[row];
    const float ri = rad[row];
    // |pi - pj|^2 with margin against f32 cancellation (coords ~1e3)
    const float d20 = ni + nj0 - 2.0f * D0[v];
    const float d21 = ni + nj1 - 2.0f * D1[v];
    const float t0 = ri + rj0, t1 = ri + rj1;
    pmask |= (d20 <= t0 * t0 + 16.0f ? 1u : 0u) << v;
    pmask |= (d21 <= t1 * t1 + 16.0f ? 1u : 0u) << (v + 8);
  }

  // --- exact clip only for surviving pairs (divergent, compact loop) ------
  unsigned decs = 0u;
  while (pmask) {
    const int b = __builtin_ctz(pmask);
    pmask &= pmask - 1u;
    const int v    = b & 7;
    const int tile = b >> 3;
    const int row  = rbase + v;
    const int col  = tile ? col1 : col0;
    const float aj = tile ? aj1 : aj0;
    const float ai = area[row];
    const float ia = interArea(&corners[row * 8], &corners[col * 8]);
    const bool hit = ia >= THR * (ai + aj - ia + 1e-8f);
    decs |= (hit ? 1u : 0u) << b;
  }
  const unsigned dec0 = decs & 0xFFu;
  const unsigned dec1 = decs >> 8;

  // --- ballot-compact decision bits into full 32-bit mask words -----------
#pragma unroll
  for (int v = 0; v < 8; ++v) {
    const unsigned balA = __builtin_amdgcn_ballot_w32((dec0 >> v) & 1u);
    const unsigned balB = __builtin_amdgcn_ballot_w32((dec1 >> v) & 1u);
    if (lane == v)
      mask[(i0 + v) * WORDS + tj] = (balA & 0xFFFFu) | (balB << 16);
    if (lane == 16 + v)
      mask[(i0 + 8 + v) * WORDS + tj] = (balA >> 16) | (balB & 0xFFFF0000u);
  }
}

// ---------------------------------------------------------------------------
// Kernel 3: serial greedy NMS, single wave32.  2048-bit suppression mask in
// LDS (2 dwords/lane); only rows of kept boxes are read from the bitmask.
// ---------------------------------------------------------------------------
__global__ void greedy_kernel(const unsigned* __restrict__ mask,
                              int* __restrict__ keep) {
  __shared__ unsigned sup[WORDS];
  const int lane = threadIdx.x;  // 0..31
  sup[2 * lane + 0] = 0u;
  sup[2 * lane + 1] = 0u;
  __syncthreads();

  int cnt = 0;  // uniform across the wave
  for (int i = 0; i < NB; ++i) {
    const int w = i >> 5, b = i & 31;
    const unsigned s = sup[w];                       // uniform LDS broadcast
    const bool take = (((s >> b) & 1u) == 0u) && (cnt < TOPN);
    if (take) {
      if (lane == 0) keep[cnt] = i;
      // OR row i into suppression mask, only bits j > i
      const int w0 = 2 * lane, w1 = 2 * lane + 1;
      const uint2 r = *(const uint2*)&mask[i * WORDS + w0];  // B64 load
      unsigned r0 = r.x, r1 = r.y;
      // zero bits with j <= i   (note 2u<<31 == 0 -> full clear, intended)
      r0 = (w0 < w) ? 0u : (w0 == w) ? (r0 & ~((2u << b) - 1u)) : r0;
      r1 = (w1 < w) ? 0u : (w1 == w) ? (r1 & ~((2u << b) - 1u)) : r1;
      sup[w0] |= r0;
      sup[w1] |= r1;
      cnt += 1;
    }
    __syncthreads();
  }
  for (int k = cnt + lane; k < TOPN; k += 32) keep[k] = -1;
}

// ---------------------------------------------------------------------------
extern "C" void kernel_launch(void* const* d_in, const int* in_sizes, int n_in,
                              void* d_out, int out_size, void* d_ws, size_t ws_size,
                              hipStream_t stream) {
  (void)in_sizes; (void)n_in; (void)out_size; (void)ws_size;
  const float* boxes = (const float*)d_in[0];   // (2048, 5) f32
  int* keep = (int*)d_out;                      // (1000,) i32

  float* corners = (float*)d_ws;                // 2048*8
  float* nsq  = corners + NB * 8;               // 2048
  float* rad  = nsq + NB;                       // 2048
  float* area = rad + NB;                       // 2048
  unsigned* mask = (unsigned*)(area + NB);      // 2048*64 words (512 KB)

  prep_kernel<<<(NB + 255) / 256, 256, 0, stream>>>(boxes, corners, nsq, rad, area);

  // 128 i-tiles * 64 j-word-tiles = 8192 waves; 8 waves / 256-thread block
  pair_kernel<<<8192 / 8, 256, 0, stream>>>(boxes, corners, nsq, rad, area, mask);

  greedy_kernel<<<1, 32, 0, stream>>>(mask, keep);
}